// SSIMLoss_996432413303
// MI455X (gfx1250) — compile-verified
//
#include <hip/hip_runtime.h>

// ---------------------------------------------------------------------------
// SSIM (7x7 box, VALID) for B x 1 x 512 x 512 fp32, per-image mean.
// Memory-bound: 67 MB streamed once -> ~2.9 us floor at 23.3 TB/s.
// Uses CDNA5 async global->LDS loads (ASYNCcnt path) for the tile fill and
// separable rolling box sums to keep fp32 VALU work ~70 FLOPs/pixel.
// Deterministic: no float atomics; fixed-order tree reductions.
// ---------------------------------------------------------------------------

#define IMG_W   512
#define HW      (IMG_W * IMG_W)
#define WIN     7
#define OUT_W   (IMG_W - WIN + 1)   // 506
#define TILE    32                  // output tile edge
#define IN_T    (TILE + WIN - 1)    // 38 input rows per tile
#define LDS_W   40                  // padded/loaded row width (10 x float4)
#define NTILE   16                  // 16*32 = 512 covers 506 outputs
#define TPI     (NTILE * NTILE)     // tiles per image = 256
#define HITEMS  (IN_T * TILE)       // 1216 horizontal-sum items per tile

__global__ __launch_bounds__(256)
void ssim_tile_kernel(const float* __restrict__ X,
                      const float* __restrict__ Y,
                      const float* __restrict__ data_range,
                      float* __restrict__ ws)
{
    __shared__ __align__(16) float sTile[2 * IN_T * LDS_W]; // X tile, then Y tile
    __shared__ float hBuf[5 * HITEMS];                      // hX,hY,hXX,hYY,hXY
    __shared__ float red[256];

    const int tid = threadIdx.x;
    const int bx = blockIdx.x, by = blockIdx.y, b = blockIdx.z;
    const int ox = bx * TILE, oy = by * TILE;

    // Column base for the 40-wide aligned load; shift last tile so the
    // 160-byte row stays inside the image (472*4 bytes is 16B-aligned).
    int colbase = ox;
    if (colbase + LDS_W > IMG_W) colbase = IMG_W - LDS_W;
    const int colshift = ox - colbase;

    const float* xb = X + (size_t)b * HW;   // uniform -> SGPR pair
    const float* yb = Y + (size_t)b * HW;

    // ---- Phase A: async global -> LDS tile fill (CDNA5 ASYNCcnt path) ----
    const unsigned ldsBase = (unsigned)(size_t)(void*)sTile; // low 32 bits = LDS byte offset
    for (int idx = tid; idx < IN_T * (LDS_W / 4); idx += 256) {
        const int chunk = idx % (LDS_W / 4);       // which float4 in the row
        const int row   = idx / (LDS_W / 4);       // 0..37
        const int rowg  = oy + row;
        if (rowg < IMG_W) {                        // skip rows past the image
            const unsigned gOff = (unsigned)((rowg * IMG_W + colbase + chunk * 4) * 4);
            const unsigned lX   = ldsBase + (unsigned)((row * LDS_W + chunk * 4) * 4);
            const unsigned lY   = lX + (unsigned)(IN_T * LDS_W * 4);
            asm volatile("global_load_async_to_lds_b128 %0, %1, %2"
                         :: "v"(lX), "v"(gOff), "s"(xb) : "memory");
            asm volatile("global_load_async_to_lds_b128 %0, %1, %2"
                         :: "v"(lY), "v"(gOff), "s"(yb) : "memory");
        }
    }
    asm volatile("s_wait_asynccnt 0" ::: "memory");
    __syncthreads();

    float* hX  = hBuf + 0 * HITEMS;
    float* hY  = hBuf + 1 * HITEMS;
    float* hXX = hBuf + 2 * HITEMS;
    float* hYY = hBuf + 3 * HITEMS;
    float* hXY = hBuf + 4 * HITEMS;

    // ---- Phase B: horizontal rolling 7-tap sums (runs of 4 along x) ----
    // 38 rows * 8 runs = 304 runs. Out-of-range columns produce garbage that
    // is only ever written to h-slots whose j is invalid (guarded in phase C).
    for (int run = tid; run < IN_T * (TILE / 4); run += 256) {
        const int r  = run >> 3;
        const int j0 = (run & 7) << 2;
        const float* px = &sTile[r * LDS_W + j0 + colshift];
        const float* py = px + IN_T * LDS_W;
        float sx = 0.f, sy = 0.f, sxx = 0.f, syy = 0.f, sxy = 0.f;
#pragma unroll
        for (int d = 0; d < WIN; ++d) {
            const float x = px[d], y = py[d];
            sx += x; sy += y; sxx += x * x; syy += y * y; sxy += x * y;
        }
        int o = r * TILE + j0;
        hX[o] = sx; hY[o] = sy; hXX[o] = sxx; hYY[o] = syy; hXY[o] = sxy;
#pragma unroll
        for (int t = 1; t < 4; ++t) {
            const float xo = px[t - 1], yo = py[t - 1];
            const float xi = px[t + WIN - 1], yi = py[t + WIN - 1];
            sx  += xi - xo;            sy  += yi - yo;
            sxx += xi * xi - xo * xo;  syy += yi * yi - yo * yo;
            sxy += xi * yi - xo * yo;
            o = r * TILE + j0 + t;
            hX[o] = sx; hY[o] = sy; hXX[o] = sxx; hYY[o] = syy; hXY[o] = sxy;
        }
    }
    __syncthreads();

    // ---- Phase C: vertical rolling 7-tap sums + SSIM (runs of 4 along y) ----
    const float dr = data_range[b];
    float C1 = 0.01f * dr; C1 *= C1;
    float C2 = 0.03f * dr; C2 *= C2;
    const float inv49 = 1.0f / 49.0f;
    const float cov   = 49.0f / 48.0f;

    const int j  = tid & 31;
    const int i0 = (tid >> 5) << 2;  // 0,4,...,28 -> 256 runs, fully balanced
    float Sx = 0.f, Sy = 0.f, Sxx = 0.f, Syy = 0.f, Sxy = 0.f;
#pragma unroll
    for (int k = 0; k < WIN; ++k) {
        const int o = (i0 + k) * TILE + j;
        Sx += hX[o]; Sy += hY[o]; Sxx += hXX[o]; Syy += hYY[o]; Sxy += hXY[o];
    }
    float acc = 0.f;
#pragma unroll
    for (int t = 0; t < 4; ++t) {
        const int i = i0 + t;
        if (t) {
            const int oi = (i + WIN - 1) * TILE + j;
            const int oo = (i - 1) * TILE + j;
            Sx  += hX[oi]  - hX[oo];   Sy  += hY[oi]  - hY[oo];
            Sxx += hXX[oi] - hXX[oo];  Syy += hYY[oi] - hYY[oo];
            Sxy += hXY[oi] - hXY[oo];
        }
        if ((oy + i) < OUT_W && (ox + j) < OUT_W) {
            const float ux = Sx * inv49, uy = Sy * inv49;
            const float vx  = cov * (Sxx * inv49 - ux * ux);
            const float vy  = cov * (Syy * inv49 - uy * uy);
            const float vxy = cov * (Sxy * inv49 - ux * uy);
            const float A1 = 2.f * ux * uy + C1;
            const float A2 = 2.f * vxy + C2;
            const float B1 = ux * ux + uy * uy + C1;
            const float B2 = vx + vy + C2;
            acc += (A1 * A2) / (B1 * B2);
        }
    }

    // ---- Phase D: deterministic workgroup tree reduction -> per-tile sum ----
    red[tid] = acc;
    __syncthreads();
#pragma unroll
    for (int s = 128; s > 0; s >>= 1) {
        if (tid < s) red[tid] += red[tid + s];
        __syncthreads();
    }
    if (tid == 0)
        ws[(size_t)b * TPI + by * NTILE + bx] = red[0];
}

__global__ __launch_bounds__(256)
void ssim_reduce_kernel(const float* __restrict__ ws, float* __restrict__ out)
{
    __shared__ float red[256];
    const int b = blockIdx.x, tid = threadIdx.x;
    red[tid] = ws[(size_t)b * TPI + tid];
    __syncthreads();
#pragma unroll
    for (int s = 128; s > 0; s >>= 1) {
        if (tid < s) red[tid] += red[tid + s];
        __syncthreads();
    }
    if (tid == 0)
        out[b] = red[0] * (1.0f / ((float)OUT_W * (float)OUT_W));
}

extern "C" void kernel_launch(void* const* d_in, const int* in_sizes, int n_in,
                              void* d_out, int out_size, void* d_ws, size_t ws_size,
                              hipStream_t stream) {
    (void)n_in; (void)out_size; (void)ws_size;
    const float* X  = (const float*)d_in[0];
    const float* Y  = (const float*)d_in[1];
    const float* dr = (const float*)d_in[2];
    float* out = (float*)d_out;
    float* ws  = (float*)d_ws;

    const int B = in_sizes[0] / HW;   // 32 for the reference setup

    dim3 grid(NTILE, NTILE, B);
    ssim_tile_kernel<<<grid, 256, 0, stream>>>(X, Y, dr, ws);
    ssim_reduce_kernel<<<B, 256, 0, stream>>>(ws, out);
}